// Router_72026601554546
// MI455X (gfx1250) — compile-verified
//
#include <hip/hip_runtime.h>

typedef float v2f __attribute__((ext_vector_type(2)));
typedef float v8f __attribute__((ext_vector_type(8)));
typedef int   v4i __attribute__((ext_vector_type(4)));

#define TOKENS   8192
#define HDIM     4096
#define NEXPERT  64
#define BT       128     // tokens per block (4 waves x 32 tokens)
#define BK       32      // k-chunk
#define NTHREADS 128     // 4 waves (wave32)
#define NCHUNK   (HDIM / BK)
#define XCH      (BT * BK)        // x-tile floats per chunk  (4096)
#define WCH      (NEXPERT * BK)   // W-tile floats per chunk  (2048)
#define CHUNKF   (XCH + WCH)      // 6144 floats = 24KB per buffer

#if __has_builtin(__builtin_amdgcn_global_load_async_to_lds_b128) && \
    __has_builtin(__builtin_amdgcn_s_wait_asynccnt)
#define USE_ASYNC 1
typedef __attribute__((address_space(1))) v4i* gas_ptr;   // global <4 x i32>*
typedef __attribute__((address_space(3))) v4i* ldsp_t;    // LDS    <4 x i32>*
#else
#define USE_ASYNC 0
#endif

__global__ void router_zero_counts(float* __restrict__ counts) {
    counts[threadIdx.x] = 0.0f;
}

// Stage one BK-wide chunk of x (BT rows) and W (NEXPERT rows) into an LDS buffer.
__device__ __forceinline__ void stage_chunk(const float* __restrict__ x,
                                            const float* __restrict__ W,
                                            float* __restrict__ buf,
                                            size_t t0, int k0, int tid) {
    float* xb = buf;
    float* wb = buf + XCH;
    #pragma unroll
    for (int i = 0; i < XCH / (NTHREADS * 4); ++i) {   // 8 x float4 per thread
        int idx = tid + i * NTHREADS;
        int row = idx >> 3;           // BK/4 == 8 float4 per row
        int c4  = idx & 7;
        const float* gp = x + (t0 + row) * HDIM + k0 + c4 * 4;
        float*       lp = xb + row * BK + c4 * 4;
#if USE_ASYNC
        __builtin_amdgcn_global_load_async_to_lds_b128(
            (gas_ptr)(unsigned long long)gp,
            (ldsp_t)(unsigned)(unsigned long long)lp, 0, 0);
#else
        *(float4*)lp = *(const float4*)gp;
#endif
    }
    #pragma unroll
    for (int i = 0; i < WCH / (NTHREADS * 4); ++i) {   // 4 x float4 per thread
        int idx = tid + i * NTHREADS;
        int row = idx >> 3;
        int c4  = idx & 7;
        const float* gp = W + (size_t)row * HDIM + k0 + c4 * 4;
        float*       lp = wb + row * BK + c4 * 4;
#if USE_ASYNC
        __builtin_amdgcn_global_load_async_to_lds_b128(
            (gas_ptr)(unsigned long long)gp,
            (ldsp_t)(unsigned)(unsigned long long)lp, 0, 0);
#else
        *(float4*)lp = *(const float4*)gp;
#endif
    }
}

__device__ __forceinline__ void wait_stage_done() {
#if USE_ASYNC
    __builtin_amdgcn_s_wait_asynccnt(0);
#endif
}

__global__ __launch_bounds__(NTHREADS)
void router_wmma_kernel(const float* __restrict__ x,
                        const float* __restrict__ W,
                        float* __restrict__ out_w,
                        float* __restrict__ out_idx,
                        float* __restrict__ out_scores,
                        float* __restrict__ out_counts) {
    __shared__ float smem[2 * CHUNKF];   // double-buffered staging, 48KB

    const int tid   = threadIdx.x;
    const int wave  = tid >> 5;
    const int lane  = tid & 31;
    const int lhalf = lane >> 4;   // 0: lanes 0-15, 1: lanes 16-31
    const int lmod  = lane & 15;
    const size_t t0 = (size_t)blockIdx.x * BT;

    // A-frag rows for this wave's two 16-token tiles
    const int ra0 = wave * 32 + lmod;
    const int ra1 = wave * 32 + 16 + lmod;

    v8f acc[2][4] = {};   // [token tile][expert tile], 16x16 f32 each

    // pipeline prologue: stage chunk 0
    stage_chunk(x, W, smem, t0, 0, tid);
    wait_stage_done();
    __syncthreads();

    for (int c = 0; c < NCHUNK; ++c) {
        float* cur = smem + (c & 1) * CHUNKF;
        float* nxt = smem + ((c + 1) & 1) * CHUNKF;

        if (c + 1 < NCHUNK)  // issue async copies for next chunk (no VGPR staging)
            stage_chunk(x, W, nxt, t0, (c + 1) * BK, tid);

        const float* xc = cur;
        const float* wc = cur + XCH;
        #pragma unroll
        for (int kk = 0; kk < BK; kk += 4) {
            const int ko = kk + 2 * lhalf;
            // load all fragments of this K-step, then burst 8 WMMAs
            v2f a0 = *(const v2f*)(xc + ra0 * BK + ko);
            v2f a1 = *(const v2f*)(xc + ra1 * BK + ko);
            v2f b0 = *(const v2f*)(wc + ( 0 + lmod) * BK + ko);
            v2f b1 = *(const v2f*)(wc + (16 + lmod) * BK + ko);
            v2f b2 = *(const v2f*)(wc + (32 + lmod) * BK + ko);
            v2f b3 = *(const v2f*)(wc + (48 + lmod) * BK + ko);
            acc[0][0] = __builtin_amdgcn_wmma_f32_16x16x4_f32(false, a0, false, b0, (short)0, acc[0][0], false, false);
            acc[0][1] = __builtin_amdgcn_wmma_f32_16x16x4_f32(false, a0, false, b1, (short)0, acc[0][1], false, false);
            acc[0][2] = __builtin_amdgcn_wmma_f32_16x16x4_f32(false, a0, false, b2, (short)0, acc[0][2], false, false);
            acc[0][3] = __builtin_amdgcn_wmma_f32_16x16x4_f32(false, a0, false, b3, (short)0, acc[0][3], false, false);
            acc[1][0] = __builtin_amdgcn_wmma_f32_16x16x4_f32(false, a1, false, b0, (short)0, acc[1][0], false, false);
            acc[1][1] = __builtin_amdgcn_wmma_f32_16x16x4_f32(false, a1, false, b1, (short)0, acc[1][1], false, false);
            acc[1][2] = __builtin_amdgcn_wmma_f32_16x16x4_f32(false, a1, false, b2, (short)0, acc[1][2], false, false);
            acc[1][3] = __builtin_amdgcn_wmma_f32_16x16x4_f32(false, a1, false, b3, (short)0, acc[1][3], false, false);
        }

        if (c + 1 < NCHUNK)
            wait_stage_done();   // our async writes to nxt have landed in LDS
        __syncthreads();         // everyone done reading cur / writing nxt
    }

    // ---- spill logits to LDS (reuse staging), stride NEXPERT+1 vs bank conflicts ----
    float* sc = smem;   // BT*(NEXPERT+1) = 8320 floats <= 12288
    #pragma unroll
    for (int ti = 0; ti < 2; ++ti) {
        #pragma unroll
        for (int et = 0; et < 4; ++et) {
            #pragma unroll
            for (int r = 0; r < 8; ++r) {
                // C/D layout: VGPR r -> M = r (lanes 0-15) or r+8 (lanes 16-31); N = lmod
                int tl = wave * 32 + ti * 16 + r + 8 * lhalf;
                int e  = et * 16 + lmod;
                sc[tl * (NEXPERT + 1) + e] = acc[ti][et][r];
            }
        }
    }
    __syncthreads();

    // ---- softmax / top-1: one thread per token ----
    {
        const float* row = sc + tid * (NEXPERT + 1);
        const size_t t = t0 + tid;
        float m = row[0];
        int   am = 0;
        #pragma unroll
        for (int e = 1; e < NEXPERT; ++e) {
            float v = row[e];
            if (v > m) { m = v; am = e; }   // first-occurrence argmax
        }
        float sum = 0.0f;
        #pragma unroll
        for (int e = 0; e < NEXPERT; ++e) sum += __expf(row[e] - m);
        const float inv = 1.0f / sum;
        #pragma unroll
        for (int e = 0; e < NEXPERT; ++e)
            out_scores[t * NEXPERT + e] = __expf(row[e] - m) * inv;
        out_w[t]   = inv;          // max softmax score = exp(0)/sum
        out_idx[t] = (float)am;
        atomicAdd(out_counts + am, 1.0f);
    }
}

extern "C" void kernel_launch(void* const* d_in, const int* in_sizes, int n_in,
                              void* d_out, int out_size, void* d_ws, size_t ws_size,
                              hipStream_t stream) {
    const float* x = (const float*)d_in[0];
    const float* W = (const float*)d_in[1];
    float* out        = (float*)d_out;
    float* out_w      = out;                                   // [8192]
    float* out_idx    = out + TOKENS;                          // [8192] (indices as float)
    float* out_scores = out + 2 * (size_t)TOKENS;              // [8192*64]
    float* out_counts = out + 2 * (size_t)TOKENS
                            + (size_t)TOKENS * NEXPERT;        // [64]

    router_zero_counts<<<1, NEXPERT, 0, stream>>>(out_counts);
    router_wmma_kernel<<<TOKENS / BT, NTHREADS, 0, stream>>>(
        x, W, out_w, out_idx, out_scores, out_counts);
}